// GCN_4011499454552
// MI455X (gfx1250) — compile-verified
//
#include <hip/hip_runtime.h>
#include <math.h>

#define NFEAT 133
#define HID   128
#define APAD  140   // sA row stride (>= K+3, spreads b64 reads over all banks)
#define WROWS 136   // sW rows: K padded up to multiple of 4 (max K=133 -> 136)
#define WPAD  144   // sW row stride: 2-row offset = 288 ≡ 32 (mod 64) -> disjoint banks
#define TILES 8     // 16-row tiles per block (amortizes W staging)

typedef __attribute__((ext_vector_type(2))) float v2f;
typedef __attribute__((ext_vector_type(8))) float v8f;

// ---------------------------------------------------------------- utilities
__global__ void fill_kernel(float* __restrict__ p, float v, long long n) {
    long long i = (long long)blockIdx.x * blockDim.x + threadIdx.x;
    if (i < n) p[i] = v;
}

__global__ void deg_accum_kernel(const long long* __restrict__ dst,
                                 float* __restrict__ deg, int E) {
    int e = blockIdx.x * blockDim.x + threadIdx.x;
    if (e < E) atomicAdd(&deg[dst[e]], 1.0f);
}

__global__ void deg_norm_kernel(const float* __restrict__ deg,
                                float* __restrict__ dis,
                                float* __restrict__ invd, int N) {
    int i = blockIdx.x * blockDim.x + threadIdx.x;
    if (i < N) {
        float d = deg[i];
        dis[i]  = rsqrtf(d);
        invd[i] = 1.0f / d;
    }
}

__global__ void edge_norm_kernel(const long long* __restrict__ src,
                                 const long long* __restrict__ dst,
                                 const float* __restrict__ dis,
                                 float* __restrict__ norm, int E) {
    int e = blockIdx.x * blockDim.x + threadIdx.x;
    if (e < E) norm[e] = dis[src[e]] * dis[dst[e]];
}

// ---------------------------------------------------------------- GEMM (WMMA f32)
// 256-thread block (8 wave32). W is staged in LDS once (zero-padded past K),
// then the block sweeps TILES 16-row tiles of H. Inner K-loop is branch-free:
// ds_load_b64 (A frag) + 2x ds_load_b32 (B frag) + v_wmma_f32_16x16x4_f32.
__launch_bounds__(256)
__global__ void gemm_wmma_kernel(const float* __restrict__ H,
                                 const float* __restrict__ W,
                                 float* __restrict__ HW,
                                 int N, int K) {
    __shared__ float sW[WROWS * WPAD];
    __shared__ float sA[16 * APAD];

    const int tid  = threadIdx.x;
    const int lane = tid & 31;
    const int wave = tid >> 5;

    // ---- stage W (K x 128) into LDS, rows zero-padded to WROWS ----
    for (int idx = tid; idx < WROWS * HID; idx += 256) {
        int r = idx >> 7;          // /128
        int c = idx & 127;
        sW[r * WPAD + c] = (r < K) ? W[(size_t)r * HID + c] : 0.0f;
    }

    const int m      = lane & 15;
    const int hiHalf = lane >> 4;            // 0 -> K pair {0,1}; 1 -> {2,3}
    const int col    = (wave << 4) + m;
    const int kp     = (K + 3) & ~3;         // padded K bound (128 or 136)
    const int tile0  = blockIdx.x * TILES;

    for (int t = 0; t < TILES; ++t) {
        const int row0 = (tile0 + t) * 16;
        if (row0 >= N) break;

        __syncthreads();   // previous tile's sA reads done (also covers sW stage)

        // ---- stage 16 x K tile of H into LDS (zero-padded to APAD) ----
        for (int idx = tid; idx < 16 * APAD; idx += 256) {
            int mm = idx / APAD, k = idx % APAD;
            float v = 0.0f;
            int r = row0 + mm;
            if (k < K && r < N) v = H[(size_t)r * K + k];
            sA[mm * APAD + k] = v;
        }
        // prefetch next tile's rows while this tile computes
        if (tid < 16 && row0 + 16 + tid < N)
            __builtin_prefetch(&H[(size_t)(row0 + 16 + tid) * K], 0, 1);
        __syncthreads();

        v8f acc = {0.f, 0.f, 0.f, 0.f, 0.f, 0.f, 0.f, 0.f};
        const int aBase = m * APAD + (hiHalf << 1);
        const int bBase = (hiHalf << 1) * WPAD + col;

#pragma unroll 4
        for (int k0 = 0; k0 < kp; k0 += 4) {
            v2f a, b;
            a.x = sA[aBase + k0];            // H[row0+m][kk]
            a.y = sA[aBase + k0 + 1];        // H[row0+m][kk+1]  (contiguous -> b64)
            b.x = sW[bBase + k0 * WPAD];     // W[kk][col]
            b.y = sW[bBase + (k0 + 1) * WPAD];
            acc = __builtin_amdgcn_wmma_f32_16x16x4_f32(
                      false, a, false, b, (short)0, acc, false, false);
        }

        // C/D layout: VGPR j -> M=j (lanes 0-15) / M=8+j (lanes 16-31); N=lane%16
        const int mb = hiHalf * 8;
#pragma unroll
        for (int j = 0; j < 8; ++j) {
            int r = row0 + mb + j;
            if (r < N) HW[(size_t)r * HID + col] = acc[j];
        }
    }
}

// ---------------------------------------------------------------- edge scatter
// One wave per edge; each lane moves a float4 slice of the 128-wide feature.
// hw + agg (~102 MB) are L2-resident, so gather + f32 atomics stay on-die.
__launch_bounds__(256)
__global__ void scatter_kernel(const long long* __restrict__ src,
                               const long long* __restrict__ dst,
                               const float* __restrict__ norm,
                               const float* __restrict__ hw,
                               float* __restrict__ agg, int E) {
    int e    = blockIdx.x * (blockDim.x >> 5) + (threadIdx.x >> 5);
    int lane = threadIdx.x & 31;
    if (e >= E) return;
    long long s = src[e];
    long long d = dst[e];
    float   nv  = norm[e];
    const float4 v = ((const float4*)(hw + (size_t)s * HID))[lane];
    float* ap = agg + (size_t)d * HID + lane * 4;
    atomicAdd(ap + 0, v.x * nv);
    atomicAdd(ap + 1, v.y * nv);
    atomicAdd(ap + 2, v.z * nv);
    atomicAdd(ap + 3, v.w * nv);
}

// ---------------------------------------------------------------- finalize
__global__ void finalize_kernel(const float* __restrict__ agg,
                                const float* __restrict__ hw,
                                const float* __restrict__ invd,
                                const float* __restrict__ bias,
                                float* __restrict__ out, long long total) {
    long long i = (long long)blockIdx.x * blockDim.x + threadIdx.x;
    if (i < total) {
        int node = (int)(i >> 7);    // /128
        int f    = (int)(i & 127);
        out[i] = tanhf(agg[i] + hw[i] * invd[node] + bias[f]);
    }
}

// ---------------------------------------------------------------- launch
extern "C" void kernel_launch(void* const* d_in, const int* in_sizes, int n_in,
                              void* d_out, int out_size, void* d_ws, size_t ws_size,
                              hipStream_t stream) {
    const float*     x   = (const float*)d_in[0];
    const long long* ei  = (const long long*)d_in[1];
    const float* Wl[4]   = {(const float*)d_in[2], (const float*)d_in[4],
                            (const float*)d_in[6], (const float*)d_in[8]};
    const float* Bl[4]   = {(const float*)d_in[3], (const float*)d_in[5],
                            (const float*)d_in[7], (const float*)d_in[9]};

    const int N = in_sizes[0] / NFEAT;
    const int E = in_sizes[1] / 2;
    const long long* src = ei;       // edge_index[0]
    const long long* dst = ei + E;   // edge_index[1]
    float* out = (float*)d_out;

    // workspace carve-out (256B aligned)
    char* ws = (char*)d_ws;
    size_t off = 0;
    auto carve = [&](size_t bytes) -> float* {
        float* p = (float*)(ws + off);
        off = (off + bytes + 255) & ~(size_t)255;
        return p;
    };
    float* deg  = carve((size_t)N * 4);
    float* dis  = carve((size_t)N * 4);
    float* invd = carve((size_t)N * 4);
    float* norm = carve((size_t)E * 4);
    float* hw   = carve((size_t)N * HID * 4);
    float* agg  = carve((size_t)N * HID * 4);
    float* hbuf = carve((size_t)N * HID * 4);

    const long long tot = (long long)N * HID;
    const int gN    = (N + 255) / 256;
    const int gE    = (E + 255) / 256;
    const int gTot  = (int)((tot + 255) / 256);
    const int nTile = (N + 15) / 16;
    const int gGemm = (nTile + TILES - 1) / TILES;
    const int gScat = (E + 7) / 8;      // 8 edges (waves) per 256-thread block

    // degree / normalization precompute (shared by all 4 layers)
    fill_kernel<<<gN, 256, 0, stream>>>(deg, 1.0f, N);   // self-loop
    deg_accum_kernel<<<gE, 256, 0, stream>>>(dst, deg, E);
    deg_norm_kernel<<<gN, 256, 0, stream>>>(deg, dis, invd, N);
    edge_norm_kernel<<<gE, 256, 0, stream>>>(src, dst, dis, norm, E);

    auto layer = [&](const float* hin, int K, const float* Wp, const float* bp,
                     float* hout) {
        gemm_wmma_kernel<<<gGemm, 256, 0, stream>>>(hin, Wp, hw, N, K);
        fill_kernel<<<gTot, 256, 0, stream>>>(agg, 0.0f, tot);
        scatter_kernel<<<gScat, 256, 0, stream>>>(src, dst, norm, hw, agg, E);
        finalize_kernel<<<gTot, 256, 0, stream>>>(agg, hw, invd, bp, hout, tot);
    };

    layer(x,    NFEAT, Wl[0], Bl[0], hbuf);
    layer(hbuf, HID,   Wl[1], Bl[1], out);
    layer(out,  HID,   Wl[2], Bl[2], hbuf);
    layer(hbuf, HID,   Wl[3], Bl[3], out);
}